// Centroid_59760174957097
// MI455X (gfx1250) — compile-verified
//
#include <hip/hip_runtime.h>
#include <stddef.h>
#include <stdint.h>

#define NUM_CLASSES 1000
#define EMBED_DIM   1024
#define BATCH       32768
#define FACTOR      0.3f

#define WSLICE   8                    // floats of EMBED_DIM per block (column slice)
#define THREADS  256
#define CHUNKS   4                    // batch split
#define ROWS     (BATCH / CHUNKS)     // 8192 rows per chunk

typedef float float4v __attribute__((ext_vector_type(4)));
typedef int   v4i     __attribute__((vector_size(16)));   // matches builtin param

typedef __attribute__((address_space(1))) v4i* g_v4i_ptr;  // global AS pointer
typedef __attribute__((address_space(3))) v4i* l_v4i_ptr;  // LDS AS pointer

// ---- gfx1250 async-LDS path (probe-confirmed present on this toolchain) ----
#if defined(__has_builtin)
#if __has_builtin(__builtin_amdgcn_global_load_async_to_lds_b128)
#define HAVE_ASYNC_LDS 1
#endif
#endif
#ifndef HAVE_ASYNC_LDS
#define HAVE_ASYNC_LDS 0
#endif

#if defined(__has_builtin)
#if __has_builtin(__builtin_amdgcn_s_wait_asynccnt)
#define WAIT_ASYNC0() __builtin_amdgcn_s_wait_asynccnt(0)
#endif
#endif
#ifndef WAIT_ASYNC0
#define WAIT_ASYNC0() asm volatile("s_wait_asynccnt 0x0" ::: "memory")
#endif

// ---------------------------------------------------------------------------
// Kernel 1: per-class partial sums of a WSLICE-wide column slice.
//   grid.x = EMBED_DIM/WSLICE (128), grid.y = CHUNKS (4)
//   LDS: acc[1000*8] f32 (31.25 KB) + ybuf[8192] i32 (32 KB) = 63.25 KB
//   Embed is read exactly once overall => 128 MiB stream, ~5.8us HBM floor.
// ---------------------------------------------------------------------------
__global__ __launch_bounds__(THREADS) void centroid_accum(
    const float* __restrict__ embed, const int* __restrict__ y,
    float* __restrict__ out /* [NUM_CLASSES*EMBED_DIM] f32 sums, pre-zeroed */) {
  __shared__ float acc[NUM_CLASSES * WSLICE];
  __shared__ int   ybuf[ROWS];

  const int tid  = threadIdx.x;
  const int col0 = blockIdx.x * WSLICE;
  const int row0 = blockIdx.y * ROWS;

  // Vectorized zero of the LDS accumulator (2000 float4s).
  {
    float4v z = {0.0f, 0.0f, 0.0f, 0.0f};
    float4v* a4 = (float4v*)acc;
    for (int i = tid; i < NUM_CLASSES * WSLICE / 4; i += THREADS) a4[i] = z;
  }

#if HAVE_ASYNC_LDS
  {
    // Stage this chunk's y[] (32 KB) into LDS with async b128 DMA loads.
    const int* gsrc = y + row0;
#pragma unroll
    for (int i = 0; i < ROWS / (4 * THREADS); ++i) {   // 8 iterations
      const int g = i * THREADS + tid;                 // b128 group index
      __builtin_amdgcn_global_load_async_to_lds_b128(
          (g_v4i_ptr)(gsrc + g * 4),
          (l_v4i_ptr)(ybuf + g * 4),
          /*offset=*/0, /*cpol=*/0);
    }
    WAIT_ASYNC0();
  }
#else
  for (int i = tid; i < ROWS; i += THREADS) ybuf[i] = y[row0 + i];
#endif
  __syncthreads();

  // Lane mapping: 2 threads per row (4 floats each, b128 loads), 128 rows/iter.
  const int sub   = tid & 1;        // which float4 of the 8-wide slice
  const int rlane = tid >> 1;       // 0..127
  const float* gbase =
      embed + (size_t)row0 * EMBED_DIM + col0 + (size_t)sub * 4;

#pragma unroll 4
  for (int it = 0; it < ROWS; it += THREADS / 2) {     // 64 iterations
    const int r   = it + rlane;
    const int cls = ybuf[r];
    const float4v* gp = (const float4v*)(gbase + (size_t)r * EMBED_DIM);
    // Prefetch the stream a few iterations ahead; uniform guard keeps the
    // prefetch inside the allocation (WGP-scope prefetch is non-speculative).
    if (it + 4 * (THREADS / 2) < ROWS) {
      __builtin_prefetch(
          (const void*)(gbase + (size_t)(r + 4 * (THREADS / 2)) * EMBED_DIM), 0,
          0);
    }
    const float4v v = __builtin_nontemporal_load(gp);
    float* a = &acc[cls * WSLICE + sub * 4];
    __hip_atomic_fetch_add(a + 0, v.x, __ATOMIC_RELAXED, __HIP_MEMORY_SCOPE_WORKGROUP);
    __hip_atomic_fetch_add(a + 1, v.y, __ATOMIC_RELAXED, __HIP_MEMORY_SCOPE_WORKGROUP);
    __hip_atomic_fetch_add(a + 2, v.z, __ATOMIC_RELAXED, __HIP_MEMORY_SCOPE_WORKGROUP);
    __hip_atomic_fetch_add(a + 3, v.w, __ATOMIC_RELAXED, __HIP_MEMORY_SCOPE_WORKGROUP);
  }
  __syncthreads();

  // Flush workgroup-private partials with fp32 global atomics (L2-resident
  // 4 MB target). Vectorized LDS reads (ds_load_b128), 4 atomics per float4.
  // i indexes float4s: class = i >> 1, sub-quad = i & 1.
  const float4v* a4 = (const float4v*)acc;
  for (int i = tid; i < NUM_CLASSES * WSLICE / 4; i += THREADS) {
    const float4v v = a4[i];
    float* dst = out + (size_t)(i >> 1) * EMBED_DIM + col0 + (i & 1) * 4;
    __hip_atomic_fetch_add(dst + 0, v.x, __ATOMIC_RELAXED, __HIP_MEMORY_SCOPE_AGENT);
    __hip_atomic_fetch_add(dst + 1, v.y, __ATOMIC_RELAXED, __HIP_MEMORY_SCOPE_AGENT);
    __hip_atomic_fetch_add(dst + 2, v.z, __ATOMIC_RELAXED, __HIP_MEMORY_SCOPE_AGENT);
    __hip_atomic_fetch_add(dst + 3, v.w, __ATOMIC_RELAXED, __HIP_MEMORY_SCOPE_AGENT);
  }
}

// ---------------------------------------------------------------------------
// Kernel 2: per-class counts (LDS histogram then global atomic flush).
// ---------------------------------------------------------------------------
__global__ __launch_bounds__(THREADS) void centroid_counts(
    const int* __restrict__ y, float* __restrict__ counts /* pre-zeroed */) {
  __shared__ float h[NUM_CLASSES];
  for (int i = threadIdx.x; i < NUM_CLASSES; i += THREADS) h[i] = 0.0f;
  __syncthreads();
  const int stride = gridDim.x * THREADS;
  for (int i = blockIdx.x * THREADS + threadIdx.x; i < BATCH; i += stride) {
    __hip_atomic_fetch_add(&h[y[i]], 1.0f, __ATOMIC_RELAXED,
                           __HIP_MEMORY_SCOPE_WORKGROUP);
  }
  __syncthreads();
  for (int i = threadIdx.x; i < NUM_CLASSES; i += THREADS) {
    const float v = h[i];
    if (v != 0.0f)
      __hip_atomic_fetch_add(&counts[i], v, __ATOMIC_RELAXED,
                             __HIP_MEMORY_SCOPE_AGENT);
  }
}

// ---------------------------------------------------------------------------
// Kernel 3: out = FACTOR * out / counts + (1-FACTOR) * centroid (in place).
//   grid.x = NUM_CLASSES, float4 vectorized: 256 threads * 4 = 1024 cols.
// ---------------------------------------------------------------------------
__global__ __launch_bounds__(THREADS) void centroid_finalize(
    const float* __restrict__ centroid, const float* __restrict__ counts,
    float* __restrict__ out) {
  const int c = blockIdx.x;
  const float inv = FACTOR / counts[c];
  const size_t base = (size_t)c * EMBED_DIM;
  float4v* o4 = (float4v*)(out + base);
  const float4v* c4 = (const float4v*)(centroid + base);
  const int i = threadIdx.x;            // EMBED_DIM/4 == THREADS exactly
  float4v v = o4[i];
  const float4v ce = c4[i];
  v = v * inv + (1.0f - FACTOR) * ce;
  o4[i] = v;
}

// ---------------------------------------------------------------------------
extern "C" void kernel_launch(void* const* d_in, const int* in_sizes, int n_in,
                              void* d_out, int out_size, void* d_ws,
                              size_t ws_size, hipStream_t stream) {
  const float* embed    = (const float*)d_in[0];   // [BATCH, EMBED_DIM] f32
  const int*   y        = (const int*)d_in[1];     // [BATCH] int
  const float* centroid = (const float*)d_in[2];   // [NUM_CLASSES, EMBED_DIM] f32
  float* out    = (float*)d_out;                   // [NUM_CLASSES, EMBED_DIM] f32
  float* counts = (float*)d_ws;                    // [NUM_CLASSES] f32 scratch

  (void)hipMemsetAsync(d_out, 0, sizeof(float) * NUM_CLASSES * EMBED_DIM,
                       stream);
  (void)hipMemsetAsync(counts, 0, sizeof(float) * NUM_CLASSES, stream);

  centroid_accum<<<dim3(EMBED_DIM / WSLICE, CHUNKS), THREADS, 0, stream>>>(
      embed, y, out);
  centroid_counts<<<dim3(32), THREADS, 0, stream>>>(y, counts);
  centroid_finalize<<<dim3(NUM_CLASSES), THREADS, 0, stream>>>(centroid, counts,
                                                               out);
}